// PhysDecoder_43044162240769
// MI455X (gfx1250) — compile-verified
//
#include <hip/hip_runtime.h>
#include <math.h>

typedef float v2f __attribute__((ext_vector_type(2)));
typedef float v8f __attribute__((ext_vector_type(8)));

#define XS   256
#define IMGN 64
#define NPTS 65536
#define IMG_ELEMS (XS * XS)                 // 65536
#define ALL_IMG_ELEMS (IMGN * IMG_ELEMS)    // 4194304

// ---------------------------------------------------------------------------
// Zero the image region of d_out (poisoned between runs; atomics need zeros).
// ---------------------------------------------------------------------------
__global__ void zero_f32(float* __restrict__ p) {
    size_t i = (size_t)blockIdx.x * blockDim.x + threadIdx.x;   // float4 index
    float4 z = {0.f, 0.f, 0.f, 0.f};
    ((float4*)p)[i] = z;
}

// ---------------------------------------------------------------------------
// Point transform + bilinear scatter-add (JAX drops OOB scatter indices).
// One thread per (b, n). rotations/shifts are block-uniform -> scalar loads.
// ---------------------------------------------------------------------------
__global__ void scatter_kernel(const float* __restrict__ flow,
                               const float* __restrict__ coords,
                               const float* __restrict__ values,
                               const float* __restrict__ rot,
                               const float* __restrict__ shifts,
                               float* __restrict__ images) {
    const float factor = 0.5f * (float)XS;   // 128
    int b = blockIdx.y;
    int n = blockIdx.x * blockDim.x + threadIdx.x;

    const float* __restrict__ fl = flow + ((size_t)b * NPTS + n) * 3;
    float px = factor * coords[n * 3 + 0] + fl[0];
    float py = factor * coords[n * 3 + 1] + fl[1];
    float pz = factor * coords[n * 3 + 2] + fl[2];

    const float* __restrict__ R = rot + (size_t)b * 9;
    float r0 = px * R[0] + py * R[1] + pz * R[2];
    float r1 = px * R[3] + py * R[4] + pz * R[5];

    float c0 = r0 - shifts[b * 2 + 0] + factor;
    float c1 = r1 - shifts[b * 2 + 1] + factor;
    // cs = (c[...,1], c[...,0]) : row coord = c1, col coord = c0
    float cs0 = c1, cs1 = c0;

    float fl0 = floorf(cs0), fl1 = floorf(cs1);
    int   i0 = (int)fl0,     i1 = (int)fl1;
    float f0 = cs0 - fl0,    f1 = cs1 - fl1;

    float v   = values[n];
    float w00 = v * (1.f - f0) * (1.f - f1);  // (i0  , i1  )
    float w10 = v * f0 * (1.f - f1);          // (i0+1, i1  )
    float w11 = v * f0 * f1;                  // (i0+1, i1+1)
    float w01 = v * (1.f - f0) * f1;          // (i0  , i1+1)

    float* __restrict__ img = images + (size_t)b * IMG_ELEMS;
    if ((unsigned)i0       < (unsigned)XS && (unsigned)i1       < (unsigned)XS) atomicAdd(img +  i0      * XS + i1,     w00);
    if ((unsigned)(i0 + 1) < (unsigned)XS && (unsigned)i1       < (unsigned)XS) atomicAdd(img + (i0 + 1) * XS + i1,     w10);
    if ((unsigned)(i0 + 1) < (unsigned)XS && (unsigned)(i1 + 1) < (unsigned)XS) atomicAdd(img + (i0 + 1) * XS + i1 + 1, w11);
    if ((unsigned)i0       < (unsigned)XS && (unsigned)(i1 + 1) < (unsigned)XS) atomicAdd(img +  i0      * XS + i1 + 1, w01);
}

// ---------------------------------------------------------------------------
// One blur pass: Y = W * X with W[r,s] = k[s-r] (banded, symmetric, 9 taps).
// Output stored TRANSPOSED so two identical passes implement W * I * W with
// the final image back in natural orientation.
//
// Block = 32x8 (8 waves). Each wave computes one 16x16 output tile via
// v_wmma_f32_16x16x4_f32 chains (3 aligned K-tiles of the band x 4 k-steps).
// A operand is synthesized from taps in LDS (branchless via padded table);
// B operand streams rows of X (fully coalesced 16-float row segments).
//
// The band boundary guard is forced into an SGPR via readfirstlane (it is
// wave-uniform at runtime since blockDim.x == 32) so the compiler emits a
// scalar branch instead of EXEC masking: WMMA requires EXEC == all-ones.
//
// WMMA f32 layouts (ISA 7.12.2, wave32):
//   A 16x4 : lane half h, m = lane&15 ; VGPR0 -> K=2h, VGPR1 -> K=2h+1
//   B 4x16 : lane half h, n = lane&15 ; VGPR0 -> K=2h, VGPR1 -> K=2h+1
//   C/D    : VGPR v, lanes 0-15 -> (M=v, N=lane), lanes 16-31 -> (M=v+8)
// ---------------------------------------------------------------------------
__global__ void blur_pass_wmma(const float* __restrict__ src,
                               float* __restrict__ dst,
                               const float* __restrict__ sigma_p) {
    __shared__ float s_ext[64];   // taps at [31-4 .. 31+4], zero elsewhere

    int lt = threadIdx.y * 32 + threadIdx.x;
    if (lt < 64) s_ext[lt] = 0.0f;
    __syncthreads();
    if (lt == 0) {
        float sig = sigma_p[0];
        float inv = -1.0f / (2.0f * sig * sig);
        float w[9], s = 0.0f;
        #pragma unroll
        for (int t = -4; t <= 4; ++t) { float e = expf((float)(t * t) * inv); w[t + 4] = e; s += e; }
        float rs = 1.0f / s;
        #pragma unroll
        for (int t = 0; t < 9; ++t) s_ext[27 + t] = w[t] * rs;
    }
    __syncthreads();

    int tile = blockIdx.x * 8 + threadIdx.y;    // 16384 tiles total
    int ct   = tile & 15;                        // column tile
    int rt   = (tile >> 4) & 15;                 // row tile
    int img  = tile >> 8;                        // image index
    int r0 = rt * 16, c0 = ct * 16;

    const float* __restrict__ S = src + (size_t)img * IMG_ELEMS;
    float*       __restrict__ D = dst + (size_t)img * IMG_ELEMS;

    int lane = threadIdx.x;
    int half = lane >> 4;        // K-pair selector
    int ml   = lane & 15;        // M for A, N for B

    v8f acc = {0.f, 0.f, 0.f, 0.f, 0.f, 0.f, 0.f, 0.f};

    #pragma unroll
    for (int d = -1; d <= 1; ++d) {            // 3 aligned K-tiles of the band
        int s0 = r0 + 16 * d;
        // Wave-uniform guard (threadIdx.y is constant per wave); readfirstlane
        // puts it in an SGPR -> scalar branch, EXEC stays all-ones for WMMA.
        int ok = __builtin_amdgcn_readfirstlane(
                     (s0 >= 0 && s0 <= XS - 16) ? 1 : 0);
        if (ok) {                              // SAME zero-padding
            #pragma unroll
            for (int kk = 0; kk < 4; ++kk) {   // K = 16 in steps of 4
                int kbase = kk * 4 + 2 * half; // K of this lane's VGPR0 element
                v2f a, bv;
                int idx = 16 * d + kbase - ml; // tap offset: k[s - r]
                a.x = s_ext[31 + idx];
                a.y = s_ext[31 + idx + 1];
                int srow = s0 + kbase;
                bv.x = S[srow * XS + c0 + ml];
                bv.y = S[(srow + 1) * XS + c0 + ml];
                acc = __builtin_amdgcn_wmma_f32_16x16x4_f32(
                          false, a, false, bv, (short)0, acc, false, false);
            }
        }
    }

    // Transposed store: dst[(c0+n)*XS + (r0+m)], m = v + 8*half, n = ml.
    // Each lane writes 8 contiguous floats -> two aligned float4 stores.
    float* drow = D + (size_t)(c0 + ml) * XS + r0 + 8 * half;
    float4 lo = {acc[0], acc[1], acc[2], acc[3]};
    float4 hi = {acc[4], acc[5], acc[6], acc[7]};
    *(float4*)(drow)     = lo;
    *(float4*)(drow + 4) = hi;
}

// ---------------------------------------------------------------------------
// a,b = x @ W_adj + b_adj   (64x16 @ 16x2)
// ---------------------------------------------------------------------------
__global__ void ab_kernel(const float* __restrict__ x,
                          const float* __restrict__ W,
                          const float* __restrict__ bb,
                          float* __restrict__ out_a,
                          float* __restrict__ out_b) {
    int i = threadIdx.x;          // 0..63
    float s0 = bb[0], s1 = bb[1];
    #pragma unroll
    for (int j = 0; j < 16; ++j) {
        float xv = x[i * 16 + j];
        s0 += xv * W[j * 2 + 0];
        s1 += xv * W[j * 2 + 1];
    }
    out_a[i] = s0;
    out_b[i] = s1;
}

extern "C" void kernel_launch(void* const* d_in, const int* in_sizes, int n_in,
                              void* d_out, int out_size, void* d_ws, size_t ws_size,
                              hipStream_t stream) {
    (void)in_sizes; (void)n_in; (void)out_size; (void)ws_size;

    const float* flow   = (const float*)d_in[0];
    const float* x      = (const float*)d_in[1];
    const float* coords = (const float*)d_in[2];
    const float* values = (const float*)d_in[3];
    const float* W_adj  = (const float*)d_in[4];
    const float* b_adj  = (const float*)d_in[5];
    const float* rot    = (const float*)d_in[6];
    const float* shifts = (const float*)d_in[7];
    const float* sigma  = (const float*)d_in[8];
    // d_in[9] = xsize (int, 256) — compile-time constant here.

    float* out    = (float*)d_out;
    float* images = out;                         // 64*256*256
    float* out_a  = out + ALL_IMG_ELEMS;         // 64
    float* out_b  = out + ALL_IMG_ELEMS + IMGN;  // 64
    float* T      = (float*)d_ws;                // 16 MB intermediate (transposed)

    // 1) zero image region of d_out (float4 granularity)
    zero_f32<<<dim3(ALL_IMG_ELEMS / 4 / 256), dim3(256), 0, stream>>>(images);

    // 2) transform + bilinear scatter (atomics into L2-resident images)
    scatter_kernel<<<dim3(NPTS / 256, IMGN), dim3(256), 0, stream>>>(
        flow, coords, values, rot, shifts, images);

    // 3) vertical blur  : T   = (W * I)^T
    blur_pass_wmma<<<dim3(IMGN * 256 / 8), dim3(32, 8), 0, stream>>>(images, T, sigma);
    // 4) horizontal blur: out = (W * T)^T = W * I * W
    blur_pass_wmma<<<dim3(IMGN * 256 / 8), dim3(32, 8), 0, stream>>>(T, images, sigma);

    // 5) tiny affine head
    ab_kernel<<<dim3(1), dim3(64), 0, stream>>>(x, W_adj, b_adj, out_a, out_b);
}